// SimpleGNN_77756087927188
// MI455X (gfx1250) — compile-verified
//
#include <hip/hip_runtime.h>
#include <hip/hip_bf16.h>

typedef __attribute__((ext_vector_type(2))) float v2f;
typedef __attribute__((ext_vector_type(4))) float v4f;
typedef __attribute__((ext_vector_type(8))) float v8f;

#define D_FEAT 512

// ---------------------------------------------------------------------------
// Kernel: init degree array to 1.0f (accounts for the self-loop edge).
// ---------------------------------------------------------------------------
__global__ void gcn_init_deg(float* __restrict__ deg, int N) {
    int i = blockIdx.x * blockDim.x + threadIdx.x;
    if (i < N) deg[i] = 1.0f;
}

// ---------------------------------------------------------------------------
// Kernel: degree counting over destination nodes (float atomics; counts are
// exactly representable well past 3.2M).
// ---------------------------------------------------------------------------
__global__ void gcn_deg(const int* __restrict__ dst, float* __restrict__ deg, int E) {
    int e = blockIdx.x * blockDim.x + threadIdx.x;
    if (e < E) atomicAdd(&deg[dst[e]], 1.0f);
}

// ---------------------------------------------------------------------------
// Kernel: h = x @ W via V_WMMA_F32_16X16X4_F32.
// One wave32 computes h for 16 consecutive rows, chaining 128 WMMA ops
// (K = 512 / 4). fp32 A layout (16x4): lanes 0-15 hold K={0,1}, lanes 16-31
// hold K={2,3} -> each lane loads one float2 of its row per step.
// fp32 B layout (4x16): only column N=0 is populated with W. The 4 W values
// per step are wave-uniform (one s_load_b128); they are placed into lanes
// 0/16 by multiplying with precomputed 0.0/1.0 lane masks, so every lane
// uses the loaded values unconditionally -> no load sinking, no EXEC
// branches, pure VALU construction of the B operand.
// Column 0 of the 16x16 C result lives in lane 0 (M=0..7 in c[0..7]) and
// lane 16 (M=8..15).
// ---------------------------------------------------------------------------
__global__ void __launch_bounds__(256)
gcn_xw_wmma(const float* __restrict__ x, const float* __restrict__ W,
            float* __restrict__ h, int N) {
    int wave = (blockIdx.x * blockDim.x + threadIdx.x) >> 5;
    int lane = threadIdx.x & 31;
    int row0 = wave * 16;
    if (row0 >= N) return;              // uniform per-wave: EXEC stays all-1s

    int r    = lane & 15;               // row within the 16-row tile
    int half = lane >> 4;               // 0: K pair {0,1}, 1: K pair {2,3}

    // Lane masks: m0 selects lane 0 (K pair {0,1}), m1 selects lane 16.
    float m0 = (lane == 0)  ? 1.0f : 0.0f;
    float m1 = (lane == 16) ? 1.0f : 0.0f;

    const v2f* xv = (const v2f*)(x + (size_t)(row0 + r) * D_FEAT) + half;
    const v4f* Wv = (const v4f*)W;      // wave-uniform pointer -> s_load_b128

    v8f c = {0.f, 0.f, 0.f, 0.f, 0.f, 0.f, 0.f, 0.f};

#pragma unroll 4
    for (int k4 = 0; k4 < D_FEAT / 4; ++k4) {
        v2f a = xv[2 * k4];             // 8B per lane, contiguous in K
        v4f w = Wv[k4];                 // uniform scalar fetch of W[4k..4k+3]
        // B column N=0: lane 0 <- {w.x, w.y}, lane 16 <- {w.z, w.w}.
        v2f b;
        b.x = w.x * m0 + w.z * m1;
        b.y = w.y * m0 + w.w * m1;
        c = __builtin_amdgcn_wmma_f32_16x16x4_f32(
                /*neg_a=*/false, a, /*neg_b=*/false, b,
                /*c_mod=*/(short)0, c, /*reuse_a=*/false, /*reuse_b=*/false);
    }

    if (r == 0) {                       // lanes 0 and 16 hold column 0 of C
        float* hp = h + row0 + 8 * half;
#pragma unroll
        for (int i = 0; i < 8; ++i) hp[i] = c[i];
    }
}

// ---------------------------------------------------------------------------
// Kernel: dinv = rsqrt(deg)  (deg >= 1 always, self-loop included).
// ---------------------------------------------------------------------------
__global__ void gcn_dinv(const float* __restrict__ deg, float* __restrict__ dinv, int N) {
    int i = blockIdx.x * blockDim.x + threadIdx.x;
    if (i < N) dinv[i] = __frsqrt_rn(deg[i]);
}

// ---------------------------------------------------------------------------
// Kernel: initialize out with the self-loop contribution dinv[i]^2 * h[i].
// ---------------------------------------------------------------------------
__global__ void gcn_init_out(const float* __restrict__ dinv, const float* __restrict__ h,
                             float* __restrict__ out, int N) {
    int i = blockIdx.x * blockDim.x + threadIdx.x;
    if (i < N) {
        float di = dinv[i];
        out[i] = di * di * h[i];
    }
}

// ---------------------------------------------------------------------------
// Kernel: per-edge scatter-add. h/dinv/out are ~400 KB each -> L2-resident;
// atomics resolve in L2 atomic units, never in HBM.
// ---------------------------------------------------------------------------
__global__ void gcn_edge_aggregate(const int* __restrict__ src, const int* __restrict__ dst,
                                   const float* __restrict__ dinv, const float* __restrict__ h,
                                   float* __restrict__ out, int E) {
    int e = blockIdx.x * blockDim.x + threadIdx.x;
    if (e < E) {
        int s = src[e];
        int d = dst[e];
        atomicAdd(&out[d], dinv[s] * dinv[d] * h[s]);
    }
}

// ---------------------------------------------------------------------------
// Host-side launcher.
// d_in[0] = x        (N * 512 float)
// d_in[1] = edge_idx (2 * E int32: src = first E, dst = second E)
// d_in[2] = W        (512 float)
// d_out   = out      (N float)
// ---------------------------------------------------------------------------
extern "C" void kernel_launch(void* const* d_in, const int* in_sizes, int n_in,
                              void* d_out, int out_size, void* d_ws, size_t ws_size,
                              hipStream_t stream) {
    const float* x  = (const float*)d_in[0];
    const int*   ei = (const int*)d_in[1];
    const float* W  = (const float*)d_in[2];
    float*       out = (float*)d_out;

    const int N = out_size;                 // 100000
    const int E = in_sizes[1] / 2;          // 3200000
    const int* src = ei;
    const int* dst = ei + E;

    // Workspace layout: deg | dinv | h  (N floats each)
    float* deg  = (float*)d_ws;
    float* dinv = deg + N;
    float* h    = dinv + N;

    const int TB = 256;
    const int nBlkN = (N + TB - 1) / TB;
    const int nBlkE = (E + TB - 1) / TB;
    // One wave per 16 rows for the WMMA matvec.
    const int nWaves   = (N + 15) / 16;
    const int nBlkWmma = (nWaves * 32 + TB - 1) / TB;

    // 1. deg = 1 (self loop)
    gcn_init_deg<<<nBlkN, TB, 0, stream>>>(deg, N);
    // 2. deg += count at dst
    gcn_deg<<<nBlkE, TB, 0, stream>>>(dst, deg, E);
    // 3. h = x @ W  (WMMA, fp32 accumulate)
    gcn_xw_wmma<<<nBlkWmma, TB, 0, stream>>>(x, W, h, N);
    // 4. dinv = rsqrt(deg)
    gcn_dinv<<<nBlkN, TB, 0, stream>>>(deg, dinv, N);
    // 5. out = dinv^2 * h   (self-loop term)
    gcn_init_out<<<nBlkN, TB, 0, stream>>>(dinv, h, out, N);
    // 6. out[dst] += dinv[src]*dinv[dst]*h[src]
    gcn_edge_aggregate<<<nBlkE, TB, 0, stream>>>(src, dst, dinv, h, out, E);
}